// RPN_52596169507322
// MI455X (gfx1250) — compile-verified
//
#include <hip/hip_runtime.h>
#include <hip/hip_bf16.h>
#include <math.h>
#include <stdint.h>

// ---------------------------------------------------------------------------
// Problem constants (from reference)
// ---------------------------------------------------------------------------
#define BB      8
#define CIN     512
#define CMID    512
#define HH      64
#define WW      64
#define HW      4096            // 64*64
#define KTOT    4608            // CIN*9
#define NA      9               // anchors per location
#define NANCH   36864           // HW * NA
#define PRE_NMS 6000
#define POST_NMS 300
#define NMS_TH  0.7f
#define MIN_SZ  16.0f

typedef __attribute__((ext_vector_type(2))) float v2f;
typedef __attribute__((ext_vector_type(8))) float v8f;
typedef __attribute__((ext_vector_type(2))) int   v2i;

// ---------------------------------------------------------------------------
// Optional CDNA5 async global->LDS staging (ASYNCcnt path), compile-safe.
// Probe-derived signature: arg0 is AS(1) pointer to <2 x i32>, arg1 is the
// LDS destination (AS(3)), then immediate offset + cpol.
// ---------------------------------------------------------------------------
#if defined(__has_builtin)
#if __has_builtin(__builtin_amdgcn_global_load_async_to_lds_b64)
#define ASYNC_LDS 1
#endif
#endif

#ifdef ASYNC_LDS
typedef __attribute__((address_space(1))) v2i* gp_v2i;
typedef __attribute__((address_space(3))) v2i* lp_v2i;

__device__ __forceinline__ void async_copy_b64(const void* gsrc, void* ldst) {
    // LDS offset is the low 32 bits of the generic shared-memory address.
    __builtin_amdgcn_global_load_async_to_lds_b64(
        (gp_v2i)(uintptr_t)gsrc,
        (lp_v2i)(unsigned)(uintptr_t)ldst,
        0, 0);
}
__device__ __forceinline__ void async_drain() {
#if __has_builtin(__builtin_amdgcn_s_wait_asynccnt)
    __builtin_amdgcn_s_wait_asynccnt(0);
#else
    asm volatile("s_wait_asynccnt 0" ::: "memory");
#endif
}
#endif

// ===========================================================================
// Kernel 1: 3x3 conv (512->512) + bias + ReLU as implicit GEMM via
// V_WMMA_F32_16X16X4_F32.  M = out-channel, N = pixel, K = ci*9.
// Block = 128 threads (4 waves); block tile 128(M) x 64(N); K-chunk 72.
// Each wave computes 32(M) x 64(N): 8 WMMAs per K-step from 6 ds_load_b64.
// LDS tiles are K-pair packed (v2f) so every fragment is one b64 load.
// ===========================================================================
#define KC1 72
#define KH1 36   // KC1/2

__global__ __launch_bounds__(128) void conv3x3_wmma(
    const float* __restrict__ xin,     // [B][CIN][H][W]
    const float* __restrict__ wgt,     // [CMID][CIN][3][3]
    const float* __restrict__ bias,    // [CMID]
    float* __restrict__ xout)          // [B][CMID][H][W]
{
    __shared__ v2f As2[128][KH1 + 1];  // [m][kpair]  (pad -> 74-float stride)
    __shared__ v2f Bs2[KH1][65];       // [kpair][p]

    const int tid  = threadIdx.x;
    const int wave = tid >> 5;
    const int lane = tid & 31;
    const int l15  = lane & 15;
    const int kh0  = lane >> 4;            // selects K pair within step
    const int p0   = blockIdx.x * 64;      // pixel tile
    const int co0  = blockIdx.y * 128;     // out-channel tile
    const int b    = blockIdx.z;
    const float* xb = xin + (size_t)b * CIN * HW;

    v8f c[2][4] = {};                      // 32(M) x 64(N) per wave

    for (int k0 = 0; k0 < KTOT; k0 += KC1) {
        // ---- stage A tile (weights): 128 x 36 v2f, K-pair contiguous ------
        for (int i = tid; i < 128 * KH1; i += 128) {
            int m  = i / KH1;
            int kh = i - m * KH1;
            const float* g = wgt + (size_t)(co0 + m) * KTOT + (k0 + 2 * kh);
#ifdef ASYNC_LDS
            async_copy_b64(g, &As2[m][kh]);
#else
            As2[m][kh] = *(const v2f*)g;
#endif
        }
        // prefetch next weight chunk (global_prefetch_b8)
        if (k0 + KC1 < KTOT)
            __builtin_prefetch(wgt + (size_t)(co0 + tid) * KTOT + (k0 + KC1), 0, 3);

        // ---- stage B tile (im2col, zero-padded): 72 x 64 scalars ----------
        for (int i = tid; i < KC1 * 64; i += 128) {
            int kk = i >> 6;
            int p  = i & 63;
            int k  = k0 + kk;
            int ci = k / 9;
            int t  = k - ci * 9;
            int ky = t / 3;
            int kx = t - ky * 3;
            int pix = p0 + p;
            int y = (pix >> 6) + ky - 1;
            int x = (pix & 63) + kx - 1;
            float v = 0.0f;
            if ((unsigned)y < (unsigned)HH && (unsigned)x < (unsigned)WW)
                v = xb[(size_t)ci * HW + y * WW + x];
            ((float*)&Bs2[kk >> 1][p])[kk & 1] = v;
        }
#ifdef ASYNC_LDS
        async_drain();
#endif
        __syncthreads();

        const int ma0 = wave * 32 + l15;
        const int ma1 = ma0 + 16;
        for (int kk = 0; kk < KC1; kk += 4) {
            const int kh = (kk >> 1) + kh0;
            v2f a0 = As2[ma0][kh];
            v2f a1 = As2[ma1][kh];
            #pragma unroll
            for (int j = 0; j < 4; ++j) {
                v2f bf = Bs2[kh][j * 16 + l15];
                c[0][j] = __builtin_amdgcn_wmma_f32_16x16x4_f32(
                    false, a0, false, bf, (short)0, c[0][j], false, false);
                c[1][j] = __builtin_amdgcn_wmma_f32_16x16x4_f32(
                    false, a1, false, bf, (short)0, c[1][j], false, false);
            }
        }
        __syncthreads();
    }

    // ---- epilogue: bias + ReLU, coalesced stores (N = lane&15 contiguous) --
    const int mhalf = (lane >> 4) * 8;
    #pragma unroll
    for (int s = 0; s < 2; ++s) {
        #pragma unroll
        for (int j = 0; j < 4; ++j) {
            int pix = p0 + j * 16 + l15;
            #pragma unroll
            for (int i = 0; i < 8; ++i) {
                int co  = co0 + wave * 32 + s * 16 + i + mhalf;
                float v = c[s][j][i] + bias[co];
                v = v > 0.0f ? v : 0.0f;
                xout[(size_t)b * CMID * HW + (size_t)co * HW + pix] = v;
            }
        }
    }
}

// ===========================================================================
// Kernel 2: fused 1x1 cls (18ch) + bbox (36ch) heads, M=64 rows (54 used),
// N = 64 pixels, K = 512, K-pair packed LDS.  Epilogue does the 2-way
// softmax into d_out prob and writes bbox deltas into d_out bbox.
// ===========================================================================
#define KC2 64
#define KH2 32

__global__ __launch_bounds__(128) void heads_wmma(
    const float* __restrict__ xmid,      // [B][CMID][H][W]
    const float* __restrict__ cls_w,     // [18][CMID]
    const float* __restrict__ cls_b,     // [18]
    const float* __restrict__ bbox_w,    // [36][CMID]
    const float* __restrict__ bbox_b,    // [36]
    float* __restrict__ prob,            // [B][18][H][W]
    float* __restrict__ bbox_out)        // [B][36][H][W]
{
    __shared__ v2f As2[64][KH2 + 1];
    __shared__ v2f Bs2[KH2][65];
    __shared__ float S[64][65];          // staged outputs [channel][pixel]

    const int tid  = threadIdx.x;
    const int wave = tid >> 5;
    const int lane = tid & 31;
    const int l15  = lane & 15;
    const int kh0  = lane >> 4;
    const int p0   = blockIdx.x * 64;
    const int b    = blockIdx.z;
    const float* xb = xmid + (size_t)b * CMID * HW;

    v8f c[4] = {};

    for (int k0 = 0; k0 < CMID; k0 += KC2) {
        // A tile: rows 0-17 cls, 18-53 bbox, 54-63 zero
        for (int i = tid; i < 64 * KH2; i += 128) {
            int m  = i >> 5;        // / KH2
            int kh = i & 31;        // % KH2
            if (m < 54) {
                const float* g = (m < 18)
                    ? (cls_w  + (size_t)m * CMID        + (k0 + 2 * kh))
                    : (bbox_w + (size_t)(m - 18) * CMID + (k0 + 2 * kh));
#ifdef ASYNC_LDS
                async_copy_b64(g, &As2[m][kh]);
#else
                As2[m][kh] = *(const v2f*)g;
#endif
            } else {
                v2f z; z.x = 0.0f; z.y = 0.0f;
                As2[m][kh] = z;
            }
        }
        // B tile: direct channel rows of x
        for (int i = tid; i < KC2 * 64; i += 128) {
            int kk = i >> 6;
            int p  = i & 63;
            ((float*)&Bs2[kk >> 1][p])[kk & 1] =
                xb[(size_t)(k0 + kk) * HW + p0 + p];
        }
#ifdef ASYNC_LDS
        async_drain();
#endif
        __syncthreads();

        const int mrow = wave * 16 + l15;
        for (int kk = 0; kk < KC2; kk += 4) {
            const int kh = (kk >> 1) + kh0;
            v2f a = As2[mrow][kh];
            #pragma unroll
            for (int j = 0; j < 4; ++j) {
                v2f bf = Bs2[kh][j * 16 + l15];
                c[j] = __builtin_amdgcn_wmma_f32_16x16x4_f32(
                    false, a, false, bf, (short)0, c[j], false, false);
            }
        }
        __syncthreads();
    }

    // stage results (+bias) into LDS so softmax can pair channels a <-> a+9
    const int mhalf = (lane >> 4) * 8;
    #pragma unroll
    for (int j = 0; j < 4; ++j) {
        int p = j * 16 + l15;
        #pragma unroll
        for (int i = 0; i < 8; ++i) {
            int m = wave * 16 + i + mhalf;
            float v = c[j][i];
            if (m < 18)      v += cls_b[m];
            else if (m < 54) v += bbox_b[m - 18];
            S[m][p] = v;
        }
    }
    __syncthreads();

    // softmax over pairs (a, a+9) -> prob
    for (int i = tid; i < 18 * 64; i += 128) {
        int ch = i >> 6;
        int p  = i & 63;
        float s0 = S[ch][p];
        int other = (ch < NA) ? ch + NA : ch - NA;
        float s1 = S[other][p];
        float mx = fmaxf(s0, s1);
        float e0 = expf(s0 - mx);
        float e1 = expf(s1 - mx);
        prob[(size_t)b * 18 * HW + (size_t)ch * HW + p0 + p] = e0 / (e0 + e1);
    }
    // raw bbox deltas
    for (int i = tid; i < 36 * 64; i += 128) {
        int ch = i >> 6;
        int p  = i & 63;
        bbox_out[(size_t)b * 36 * HW + (size_t)ch * HW + p0 + p] = S[18 + ch][p];
    }
}

// ===========================================================================
// Kernel 3: analytic anchors + delta decode + clip + min-size filter.
// boxes[b][j][4], scores[b][j] with j = pix*9 + a (matches reference layout).
// ===========================================================================
__global__ __launch_bounds__(256) void decode_kernel(
    const float* __restrict__ prob,      // [B][18][HW]
    const float* __restrict__ bbox,      // [B][36][HW]
    const float* __restrict__ im_info,   // [B][3]
    float* __restrict__ boxes,           // [B][NANCH][4]
    float* __restrict__ scores)          // [B][NANCH]
{
    int idx = blockIdx.x * 256 + threadIdx.x;
    if (idx >= BB * NANCH) return;
    int b   = idx / NANCH;
    int j   = idx - b * NANCH;
    int pix = j / NA;
    int a   = j - pix * NA;
    int hy  = pix >> 6;
    int wx  = pix & 63;

    const float ratios[3] = {0.5f, 1.0f, 2.0f};
    const float scales[3] = {8.0f, 16.0f, 32.0f};
    float r  = ratios[a / 3];
    float s  = scales[a % 3];
    float sq = sqrtf(r);
    float wsa = (16.0f / sq) * s;
    float hsa = (16.0f * sq) * s;
    float cx0 = wx * 16.0f + 7.5f;
    float cy0 = hy * 16.0f + 7.5f;
    float x1a = cx0 - 0.5f * (wsa - 1.0f);
    float y1a = cy0 - 0.5f * (hsa - 1.0f);
    float x2a = cx0 + 0.5f * (wsa - 1.0f);
    float y2a = cy0 + 0.5f * (hsa - 1.0f);

    float aw  = x2a - x1a + 1.0f;
    float ah  = y2a - y1a + 1.0f;
    float acx = x1a + 0.5f * aw;
    float acy = y1a + 0.5f * ah;

    size_t boff = (size_t)b * 36 * HW;
    float dx = bbox[boff + (size_t)(4 * a + 0) * HW + pix];
    float dy = bbox[boff + (size_t)(4 * a + 1) * HW + pix];
    float dw = bbox[boff + (size_t)(4 * a + 2) * HW + pix];
    float dh = bbox[boff + (size_t)(4 * a + 3) * HW + pix];

    float pcx = dx * aw + acx;
    float pcy = dy * ah + acy;
    float pw  = expf(dw) * aw;
    float ph  = expf(dh) * ah;

    float x1 = pcx - 0.5f * pw;
    float y1 = pcy - 0.5f * ph;
    float x2 = pcx + 0.5f * pw;
    float y2 = pcy + 0.5f * ph;

    float imH = im_info[b * 3 + 0];
    float imW = im_info[b * 3 + 1];
    float sc  = im_info[b * 3 + 2];
    x1 = fminf(fmaxf(x1, 0.0f), imW - 1.0f);
    y1 = fminf(fmaxf(y1, 0.0f), imH - 1.0f);
    x2 = fminf(fmaxf(x2, 0.0f), imW - 1.0f);
    y2 = fminf(fmaxf(y2, 0.0f), imH - 1.0f);

    float score = prob[(size_t)b * 18 * HW + (size_t)(NA + a) * HW + pix];
    bool ok = ((x2 - x1 + 1.0f) >= MIN_SZ * sc) && ((y2 - y1 + 1.0f) >= MIN_SZ * sc);
    if (!ok) score = -1000000000.0f;

    size_t o = (size_t)b * NANCH + j;
    boxes[o * 4 + 0] = x1;
    boxes[o * 4 + 1] = y1;
    boxes[o * 4 + 2] = x2;
    boxes[o * 4 + 3] = y2;
    scores[o] = score;
}

// ===========================================================================
// Kernel 4: exact top-PRE_NMS membership via rank counting.
// rank(i) = #{ j : s_j > s_i  ||  (s_j == s_i && j < i) }   (stable desc)
// valid[i] = rank < PRE_NMS.   NANCH is divisible by 256.
// ===========================================================================
__global__ __launch_bounds__(256) void rank_kernel(
    const float* __restrict__ scores,    // [B][NANCH]
    unsigned* __restrict__ valid)        // [B][NANCH]
{
    __shared__ float sh[256];
    int b = blockIdx.y;
    int i = blockIdx.x * 256 + threadIdx.x;
    const float* sb = scores + (size_t)b * NANCH;
    float si = sb[i];
    int cnt = 0;
    for (int t0 = 0; t0 < NANCH; t0 += 256) {
        sh[threadIdx.x] = sb[t0 + threadIdx.x];
        __syncthreads();
        #pragma unroll 8
        for (int t = 0; t < 256; ++t) {
            float sj = sh[t];
            int jj = t0 + t;
            cnt += ((sj > si) || (sj == si && jj < i)) ? 1 : 0;
        }
        __syncthreads();
    }
    valid[(size_t)b * NANCH + i] = (cnt < PRE_NMS) ? 1u : 0u;
}

// ===========================================================================
// Kernel 5: greedy NMS, one workgroup per batch, 300 serial rounds of
// (parallel argmax over valid) + (parallel IoU suppression).
// ===========================================================================
__global__ __launch_bounds__(1024) void nms_kernel(
    const float* __restrict__ boxes,     // [B][NANCH][4]
    const float* __restrict__ scores,    // [B][NANCH]
    unsigned* __restrict__ valid,        // [B][NANCH] (consumed)
    float* __restrict__ rois)            // [B][POST_NMS][5]
{
    __shared__ float sVal[1024];
    __shared__ int   sIdx[1024];
    __shared__ float pick[4];
    __shared__ int   pickIdx;

    const int b   = blockIdx.x;
    const int tid = threadIdx.x;
    const float* bb = boxes  + (size_t)b * NANCH * 4;
    const float* sc = scores + (size_t)b * NANCH;
    unsigned*    vd = valid  + (size_t)b * NANCH;

    for (int r = 0; r < POST_NMS; ++r) {
        // ---- argmax over valid (ties -> lowest index, all-invalid -> 0) ----
        float best = -INFINITY;
        int   bi   = 0x7fffffff;
        for (int j = tid; j < NANCH; j += 1024) {
            if (vd[j]) {
                float s = sc[j];
                if (s > best || (s == best && j < bi)) { best = s; bi = j; }
            }
        }
        sVal[tid] = best;
        sIdx[tid] = bi;
        __syncthreads();
        for (int off = 512; off > 0; off >>= 1) {
            if (tid < off) {
                float ov = sVal[tid + off];
                int   oi = sIdx[tid + off];
                if (ov > sVal[tid] || (ov == sVal[tid] && oi < sIdx[tid])) {
                    sVal[tid] = ov;
                    sIdx[tid] = oi;
                }
            }
            __syncthreads();
        }
        if (tid == 0) {
            int k = sIdx[0];
            if (k == 0x7fffffff) k = 0;   // matches argmax over all -inf
            pickIdx = k;
            pick[0] = bb[(size_t)k * 4 + 0];
            pick[1] = bb[(size_t)k * 4 + 1];
            pick[2] = bb[(size_t)k * 4 + 2];
            pick[3] = bb[(size_t)k * 4 + 3];
            float* ro = rois + ((size_t)b * POST_NMS + r) * 5;
            ro[0] = (float)b;
            ro[1] = pick[0]; ro[2] = pick[1]; ro[3] = pick[2]; ro[4] = pick[3];
        }
        __syncthreads();

        // ---- suppression ----
        float px1 = pick[0], py1 = pick[1], px2 = pick[2], py2 = pick[3];
        float pa  = (px2 - px1 + 1.0f) * (py2 - py1 + 1.0f);
        int   pk  = pickIdx;
        for (int j = tid; j < NANCH; j += 1024) {
            if (vd[j]) {
                float x1 = bb[(size_t)j * 4 + 0];
                float y1 = bb[(size_t)j * 4 + 1];
                float x2 = bb[(size_t)j * 4 + 2];
                float y2 = bb[(size_t)j * 4 + 3];
                float xx1 = fmaxf(px1, x1), yy1 = fmaxf(py1, y1);
                float xx2 = fminf(px2, x2), yy2 = fminf(py2, y2);
                float inter = fmaxf(0.0f, xx2 - xx1 + 1.0f) *
                              fmaxf(0.0f, yy2 - yy1 + 1.0f);
                float a2  = (x2 - x1 + 1.0f) * (y2 - y1 + 1.0f);
                float iou = inter / (pa + a2 - inter);
                if (iou > NMS_TH || j == pk) vd[j] = 0u;
            }
        }
        __syncthreads();
    }
}

// ===========================================================================
// Host-side launch
// ===========================================================================
extern "C" void kernel_launch(void* const* d_in, const int* in_sizes, int n_in,
                              void* d_out, int out_size, void* d_ws, size_t ws_size,
                              hipStream_t stream) {
    (void)in_sizes; (void)n_in; (void)out_size; (void)ws_size;

    const float* base_feature = (const float*)d_in[0];   // [8,512,64,64]
    const float* im_info      = (const float*)d_in[1];   // [8,3]
    // d_in[2] gt_boxes, d_in[3] num_boxes: unused by reference outputs
    const float* conv_w = (const float*)d_in[4];         // [512,512,3,3]
    const float* conv_b = (const float*)d_in[5];         // [512]
    const float* cls_w  = (const float*)d_in[6];         // [18,512,1,1]
    const float* cls_b  = (const float*)d_in[7];         // [18]
    const float* bbox_w = (const float*)d_in[8];         // [36,512,1,1]
    const float* bbox_b = (const float*)d_in[9];         // [36]

    // d_out layout: rois [8,300,5] | prob [8,18,64,64] | bbox [8,36,64,64]
    float* out_rois = (float*)d_out;
    float* out_prob = out_rois + (size_t)BB * POST_NMS * 5;
    float* out_bbox = out_prob + (size_t)BB * 18 * HW;

    // workspace layout
    char* ws = (char*)d_ws;
    float*    x_mid  = (float*)ws;                                   // 8*512*4096 f32 (64 MiB)
    float*    boxes  = (float*)(ws + (size_t)BB * CMID * HW * 4);    // 8*36864*4 f32
    float*    scores = boxes + (size_t)BB * NANCH * 4;               // 8*36864 f32
    unsigned* valid  = (unsigned*)(scores + (size_t)BB * NANCH);     // 8*36864 u32

    // 1) 3x3 conv + ReLU (WMMA fp32, 128x64 block tile)
    conv3x3_wmma<<<dim3(HW / 64, CMID / 128, BB), 128, 0, stream>>>(
        base_feature, conv_w, conv_b, x_mid);

    // 2) fused cls/bbox 1x1 heads + softmax (WMMA fp32)
    heads_wmma<<<dim3(HW / 64, 1, BB), 128, 0, stream>>>(
        x_mid, cls_w, cls_b, bbox_w, bbox_b, out_prob, out_bbox);

    // 3) anchor decode + clip + min-size filter
    decode_kernel<<<(BB * NANCH + 255) / 256, 256, 0, stream>>>(
        out_prob, out_bbox, im_info, boxes, scores);

    // 4) exact top-6000 membership (rank counting)
    rank_kernel<<<dim3(NANCH / 256, BB), 256, 0, stream>>>(scores, valid);

    // 5) greedy NMS -> rois
    nms_kernel<<<BB, 1024, 0, stream>>>(boxes, scores, valid, out_rois);
}